// GAT_4733053960619
// MI455X (gfx1250) — compile-verified
//
#include <hip/hip_runtime.h>

// ---------------- problem constants (match reference) ----------------
constexpr int Nn  = 50000;   // nodes
constexpr int Ne  = 800000;  // edges
constexpr int F   = 256;     // input feats
constexpr int D0  = 128;     // SAGE out (64*2)
constexpr int H1  = 2;       // GAT1 heads
constexpr int D1  = 64;      // GAT1 per-head dim
constexpr int C2  = 47;      // GAT2 out (1 head)
constexpr int C2P = 64;      // GAT2 padded cols

typedef float v2f __attribute__((ext_vector_type(2)));
typedef float v8f __attribute__((ext_vector_type(8)));

// order-preserving encode/decode for float atomicMax on unsigned
__device__ __forceinline__ unsigned enc_f(float f) {
    unsigned u = __float_as_uint(f);
    return (u & 0x80000000u) ? ~u : (u | 0x80000000u);
}
__device__ __forceinline__ float dec_f(unsigned u) {
    unsigned b = (u & 0x80000000u) ? (u ^ 0x80000000u) : ~u;
    return __uint_as_float(b);
}

// ---------------- weight transpose + zero-pad ----------------
// Wt[c*K + k] = (c < N) ? W[k*N + c] : 0   for c in [0, Npad)
__global__ void k_transpose(const float* __restrict__ W, float* __restrict__ Wt,
                            int K, int N, int Npad) {
    int i = blockIdx.x * blockDim.x + threadIdx.x;
    if (i >= Npad * K) return;
    const int c = i / K, k = i % K;
    Wt[i] = (c < N) ? W[(long long)k * N + c] : 0.0f;
}

// ---------------- fp32 WMMA GEMM, LDS-staged B panel ----------------
// out[M,Nout] = act( A[M,K]@W + (A2/max(deg,1))@W2 + bias ), weights pre-transposed+padded.
// Block = 8 waves sharing ONE 64-column weight panel staged in LDS (stride K+4 ->
// bank-conflict-free ds_load_b64). Each wave computes a 16x64 tile with
// 4 v_wmma_f32_16x16x4_f32 accumulators reusing the A fragment. K-loop has no guards.
template <bool HAS_A2, bool ACT>
__global__ __launch_bounds__(256)
void wmma_gemm(const float* __restrict__ A, const float* __restrict__ Wt,
               const float* __restrict__ A2, const float* __restrict__ W2t,
               const float* __restrict__ rowdeg, const float* __restrict__ bias,
               float* __restrict__ out, int M, int K, int Nout, int ncg)
{
    extern __shared__ float lds[];
    const int KP = K + 4;                     // padded LDS row stride (even, banks spread)
    const int cg   = blockIdx.x % ncg;
    const int col0 = cg * 64;
    float* w1s = lds;                         // 64 * KP
    float* w2s = lds + 64 * KP;               // 64 * KP (HAS_A2 only)

    // cooperative stage of the weight panel(s); K is a power of two
    const int ksh = 31 - __clz(K);
    for (int i = threadIdx.x; i < 64 * K; i += 256) {
        const int c = i >> ksh, k = i & (K - 1);
        w1s[c * KP + k] = Wt[(long long)(col0 + c) * K + k];
        if (HAS_A2) w2s[c * KP + k] = W2t[(long long)(col0 + c) * K + k];
    }
    __syncthreads();

    const int rt = (blockIdx.x / ncg) * 8 + (threadIdx.x >> 5);  // row tile for this wave
    if (rt >= M / 16) return;                 // after the barrier: safe
    const int lane = threadIdx.x & 31;
    const int half = lane >> 4;   // 0: K pair {0,1}/rows 0..7 ; 1: K pair {2,3}/rows 8..15
    const int l    = lane & 15;
    const int row  = rt * 16 + l;             // M is an exact multiple of 16

    const float* Arow  = A + (long long)row * K + half * 2;
    const float* A2row = HAS_A2 ? (A2 + (long long)row * K + half * 2) : nullptr;
    const float inv    = HAS_A2 ? (1.0f / fmaxf(rowdeg[row], 1.0f)) : 0.0f;
    __builtin_prefetch(Arow, 0, 3);           // pull this lane's A row toward near caches

    const float* Bs[4];
    const float* B2s[4];
#pragma unroll
    for (int t = 0; t < 4; ++t) {
        Bs[t] = w1s + (t * 16 + l) * KP + half * 2;
        if (HAS_A2) B2s[t] = w2s + (t * 16 + l) * KP + half * 2;
    }

    v8f acc[4] = {};
    for (int kk = 0; kk < K; kk += 4) {
        const v2f a = *(const v2f*)(Arow + kk);                 // global_load_b64
        v2f a2;
        if (HAS_A2) {
            const v2f t2 = *(const v2f*)(A2row + kk);
            a2.x = t2.x * inv; a2.y = t2.y * inv;
        }
#pragma unroll
        for (int t = 0; t < 4; ++t) {
            const v2f b = *(const v2f*)(Bs[t] + kk);            // ds_load_b64
            acc[t] = __builtin_amdgcn_wmma_f32_16x16x4_f32(
                false, a, false, b, (short)0, acc[t], false, false);
            if (HAS_A2) {
                const v2f b2 = *(const v2f*)(B2s[t] + kk);
                acc[t] = __builtin_amdgcn_wmma_f32_16x16x4_f32(
                    false, a2, false, b2, (short)0, acc[t], false, false);
            }
        }
    }

    // store: C/D layout — VGPR v holds rows v (lanes 0-15) and v+8 (lanes 16-31), col = lane&15
#pragma unroll
    for (int t = 0; t < 4; ++t) {
        const int c = col0 + t * 16 + l;
        if (c < Nout) {
            const float bv = bias ? bias[c] : 0.0f;
#pragma unroll
            for (int v = 0; v < 8; ++v) {
                const int r = rt * 16 + v + 8 * half;
                float x = acc[t][v] + bv;
                if (ACT) x = x > 0.0f ? x : (expf(x) - 1.0f);   // ELU
                out[(long long)r * Nout + c] = x;
            }
        }
    }
}

// ---------------- elementwise / edge kernels ----------------
__global__ void fill_f32(float* p, float v, long long n) {
    long long i = (long long)blockIdx.x * blockDim.x + threadIdx.x;
    if (i < n) p[i] = v;
}
__global__ void fill_u32(unsigned* p, unsigned v, long long n) {
    long long i = (long long)blockIdx.x * blockDim.x + threadIdx.x;
    if (i < n) p[i] = v;
}

__global__ void k_deg(const int* __restrict__ dst, float* __restrict__ deg, int E) {
    int i = blockIdx.x * blockDim.x + threadIdx.x;
    if (i < E) atomicAdd(&deg[dst[i]], 1.0f);
}

// scatter-sum of x[src] rows into neigh[dst]; 4 floats per thread
__global__ void k_neigh(const float* __restrict__ x, const int* __restrict__ src,
                        const int* __restrict__ dst, float* __restrict__ neigh, int E) {
    long long tid = (long long)blockIdx.x * blockDim.x + threadIdx.x;
    if (tid >= (long long)E * (F / 4)) return;
    const int e = (int)(tid >> 6);
    const int c = ((int)tid & 63) * 4;
    const float4 v = *(const float4*)(x + (long long)src[e] * F + c);
    float* o = neigh + (long long)dst[e] * F + c;
    atomicAdd(o + 0, v.x); atomicAdd(o + 1, v.y);
    atomicAdd(o + 2, v.z); atomicAdd(o + 3, v.w);
}

// el[n,h] = feat[n,h,:].attn_l[h,:] ; er likewise
__global__ void k_attn(const float* __restrict__ feat, const float* __restrict__ al,
                       const float* __restrict__ ar, float* __restrict__ el,
                       float* __restrict__ er, int N, int H, int D) {
    int i = blockIdx.x * blockDim.x + threadIdx.x;
    if (i >= N * H) return;
    const int n = i / H, h = i % H;
    const float* f = feat + (long long)n * H * D + h * D;
    float a = 0.0f, b = 0.0f;
    for (int d = 0; d < D; ++d) { a += f[d] * al[h * D + d]; b += f[d] * ar[h * D + d]; }
    el[i] = a; er[i] = b;
}

// pass A: e = leaky_relu(el[src]+er[dst]); segment max via encoded atomicMax
__global__ void k_edge_logit(const int* __restrict__ src, const int* __restrict__ dst,
                             const float* __restrict__ el, const float* __restrict__ er,
                             float* __restrict__ ee, unsigned* __restrict__ mx, int E, int H) {
    int i = blockIdx.x * blockDim.x + threadIdx.x;
    if (i >= E * H) return;
    const int e = i / H, h = i % H;
    float v = el[src[e] * H + h] + er[dst[e] * H + h];
    v = v > 0.0f ? v : 0.2f * v;
    ee[i] = v;
    atomicMax(&mx[dst[e] * H + h], enc_f(v));
}

// pass B: ex = exp(e - m[dst]); segment sum
__global__ void k_edge_exp(const int* __restrict__ dst, const float* __restrict__ ee,
                           const unsigned* __restrict__ mx, float* __restrict__ exe,
                           float* __restrict__ s, int E, int H) {
    int i = blockIdx.x * blockDim.x + threadIdx.x;
    if (i >= E * H) return;
    const int e = i / H, h = i % H;
    const float m = dec_f(mx[dst[e] * H + h]);
    const float ex = __expf(ee[i] - m);
    exe[i] = ex;
    atomicAdd(&s[dst[e] * H + h], ex);
}

// pass C: out[dst,h,:] += (ex / max(s[dst,h],1e-9)) * feat[src,h,:]
__global__ void k_edge_msg(const int* __restrict__ src, const int* __restrict__ dst,
                           const float* __restrict__ feat, const float* __restrict__ exe,
                           const float* __restrict__ s, float* __restrict__ out,
                           int E, int H, int D) {
    long long i = (long long)blockIdx.x * blockDim.x + threadIdx.x;
    const long long tot = (long long)E * H * D;
    if (i >= tot) return;
    const int d = (int)(i % D);
    const long long eh = i / D;
    const int h = (int)(eh % H);
    const int e = (int)(eh / H);
    const int sd = dst[e];
    const float alpha = exe[e * H + h] / fmaxf(s[sd * H + h], 1e-9f);
    atomicAdd(&out[(long long)sd * H * D + h * D + d],
              alpha * feat[(long long)src[e] * H * D + h * D + d]);
}

// out = act(in + bias[col])
__global__ void k_bias_act(const float* __restrict__ in, const float* __restrict__ b,
                           float* __restrict__ out, long long total, int Dtot, int act) {
    long long i = (long long)blockIdx.x * blockDim.x + threadIdx.x;
    if (i >= total) return;
    const int c = (int)(i % Dtot);
    float v = in[i] + b[c];
    if (act) v = v > 0.0f ? v : (expf(v) - 1.0f);
    out[i] = v;
}

// ---------------- host-side pipeline ----------------
static inline unsigned gsz(long long n, int b) { return (unsigned)((n + b - 1) / b); }

extern "C" void kernel_launch(void* const* d_in, const int* in_sizes, int n_in,
                              void* d_out, int out_size, void* d_ws, size_t ws_size,
                              hipStream_t stream) {
    const float* x       = (const float*)d_in[0];
    const int*   src     = (const int*)  d_in[1];
    const int*   dst     = (const int*)  d_in[2];
    const float* w_self  = (const float*)d_in[3];
    const float* w_neigh = (const float*)d_in[4];
    const float* sage_b  = (const float*)d_in[5];
    const float* g1_w    = (const float*)d_in[6];
    const float* g1_al   = (const float*)d_in[7];
    const float* g1_ar   = (const float*)d_in[8];
    const float* g1_b    = (const float*)d_in[9];
    const float* g2_w    = (const float*)d_in[10];
    const float* g2_al   = (const float*)d_in[11];
    const float* g2_ar   = (const float*)d_in[12];
    const float* g2_b    = (const float*)d_in[13];

    // workspace layout (floats)
    float*    ws    = (float*)d_ws;
    float*    deg   = ws;                               // Nn
    float*    neigh = deg + Nn;                         // Nn*256 ; reused after SAGE
    float*    h0    = neigh + (long long)Nn * F;        // Nn*128 ; reused as h1
    float*    el    = h0 + (long long)Nn * D0;          // Nn*2
    float*    er    = el + Nn * H1;                     // Nn*2
    unsigned* mx    = (unsigned*)(er + Nn * H1);        // Nn*2
    float*    ssum  = (float*)(mx + Nn * H1);           // Nn*2
    float*    ee    = ssum + Nn * H1;                   // Ne*2
    float*    exe   = ee + (long long)Ne * H1;          // Ne*2
    float*    wt_self  = exe + (long long)Ne * H1;      // 128*256 transposed
    float*    wt_neigh = wt_self + D0 * F;              // 128*256
    float*    wt_g1    = wt_neigh + D0 * F;             // 128*128
    float*    wt_g2    = wt_g1 + (H1 * D1) * D0;        // 64*128 (padded from 47)
    // aliases into the (dead after SAGE) neigh region:
    float*    feat  = neigh;                            // Nn*128 (GAT1 feat; later GAT2 feat Nn*47)
    float*    aggr  = neigh + (long long)Nn * D0;       // Nn*128 (GAT aggregation)

    const int B = 256;
    const unsigned NEG_INF_ENC = 0x007FFFFFu;           // enc_f(-inf)
    const int Mtiles = Nn / 16;                         // 3125
    const unsigned rowblk = gsz(Mtiles, 8);             // 8 waves (row tiles) per block

    // ---- weight transposes (pad gat2 to 64 cols with zeros) ----
    k_transpose<<<gsz(D0 * F, B), B, 0, stream>>>(w_self,  wt_self,  F,  D0, D0);
    k_transpose<<<gsz(D0 * F, B), B, 0, stream>>>(w_neigh, wt_neigh, F,  D0, D0);
    k_transpose<<<gsz(D0 * D0, B), B, 0, stream>>>(g1_w,   wt_g1,    D0, H1 * D1, H1 * D1);
    k_transpose<<<gsz(C2P * D0, B), B, 0, stream>>>(g2_w,  wt_g2,    D0, C2, C2P);

    // ---- layer 0: SAGEConv(mean) ----
    fill_f32<<<gsz(Nn, B), B, 0, stream>>>(deg, 0.0f, Nn);
    fill_f32<<<gsz((long long)Nn * F, B), B, 0, stream>>>(neigh, 0.0f, (long long)Nn * F);
    k_deg<<<gsz(Ne, B), B, 0, stream>>>(dst, deg, Ne);
    k_neigh<<<gsz((long long)Ne * (F / 4), B), B, 0, stream>>>(x, src, dst, neigh, Ne);
    // h0 = elu(x@Wself + (neigh/deg)@Wneigh + b)   M=50000 K=256 N=128
    {
        const size_t shmem = 2ull * 64 * (F + 4) * sizeof(float);    // two panels
        wmma_gemm<true, true><<<rowblk * 2, B, shmem, stream>>>(
            x, wt_self, neigh, wt_neigh, deg, sage_b, h0, Nn, F, D0, 2);
    }

    // ---- layer 1: GAT heads=2 dim=64 ----
    {
        const size_t shmem = 64ull * (D0 + 4) * sizeof(float);
        wmma_gemm<false, false><<<rowblk * 2, B, shmem, stream>>>(
            h0, wt_g1, nullptr, nullptr, nullptr, nullptr, feat, Nn, D0, H1 * D1, 2);
    }
    k_attn<<<gsz(Nn * H1, B), B, 0, stream>>>(feat, g1_al, g1_ar, el, er, Nn, H1, D1);
    fill_u32<<<gsz(Nn * H1, B), B, 0, stream>>>(mx, NEG_INF_ENC, Nn * H1);
    fill_f32<<<gsz(Nn * H1, B), B, 0, stream>>>(ssum, 0.0f, Nn * H1);
    fill_f32<<<gsz((long long)Nn * D0, B), B, 0, stream>>>(aggr, 0.0f, (long long)Nn * D0);
    k_edge_logit<<<gsz(Ne * H1, B), B, 0, stream>>>(src, dst, el, er, ee, mx, Ne, H1);
    k_edge_exp<<<gsz(Ne * H1, B), B, 0, stream>>>(dst, ee, mx, exe, ssum, Ne, H1);
    k_edge_msg<<<gsz((long long)Ne * H1 * D1, B), B, 0, stream>>>(src, dst, feat, exe, ssum,
                                                                  aggr, Ne, H1, D1);
    // h1 = elu(aggr + g1_b)  -> reuse h0 buffer
    k_bias_act<<<gsz((long long)Nn * D0, B), B, 0, stream>>>(aggr, g1_b, h0,
                                                             (long long)Nn * D0, D0, 1);

    // ---- layer 2: GAT heads=1 dim=47 ----
    // feat2 = h1 @ g2_w   M=50000 K=128 N=47 (weights zero-padded to 64 cols)
    {
        const size_t shmem = 64ull * (D0 + 4) * sizeof(float);
        wmma_gemm<false, false><<<rowblk, B, shmem, stream>>>(
            h0, wt_g2, nullptr, nullptr, nullptr, nullptr, feat, Nn, D0, C2, 1);
    }
    k_attn<<<gsz(Nn, B), B, 0, stream>>>(feat, g2_al, g2_ar, el, er, Nn, 1, C2);
    fill_u32<<<gsz(Nn, B), B, 0, stream>>>(mx, NEG_INF_ENC, Nn);
    fill_f32<<<gsz(Nn, B), B, 0, stream>>>(ssum, 0.0f, Nn);
    fill_f32<<<gsz((long long)Nn * C2, B), B, 0, stream>>>(aggr, 0.0f, (long long)Nn * C2);
    k_edge_logit<<<gsz(Ne, B), B, 0, stream>>>(src, dst, el, er, ee, mx, Ne, 1);
    k_edge_exp<<<gsz(Ne, B), B, 0, stream>>>(dst, ee, mx, exe, ssum, Ne, 1);
    k_edge_msg<<<gsz((long long)Ne * C2, B), B, 0, stream>>>(src, dst, feat, exe, ssum,
                                                             aggr, Ne, 1, C2);
    // d_out = aggr + g2_b  (mean over 1 head == identity)
    k_bias_act<<<gsz((long long)Nn * C2, B), B, 0, stream>>>(aggr, g2_b, (float*)d_out,
                                                             (long long)Nn * C2, C2, 0);
}